// ReadHead_39170101739974
// MI455X (gfx1250) — compile-verified
//
#include <hip/hip_runtime.h>
#include <math.h>

// NTM read head: B=512, C=256, W=64, N=4096.
// Memory-bound: memory_M (512 MiB) streamed twice (dot/norm pass + readout
// pass) => ~1 GiB HBM traffic => ~46us floor at 23.3 TB/s. FC layer done with
// V_WMMA_F32_16X16X4_F32 (exact f32 matrix path on CDNA5).

#define B_   512
#define C_   256
#define W_   64
#define N_   4096
#define EPS_ 1e-16f
#define OPAD 80   // FC output columns padded 70 -> 80 (5 WMMA col-tiles)

typedef __attribute__((ext_vector_type(2))) float v2f;
typedef __attribute__((ext_vector_type(8))) float v8f;

__device__ __forceinline__ float softplus_f(float x) {
    return (x > 20.f) ? x : log1pf(expf(x));
}

// ---------------------------------------------------------------------------
// Kernel 1: out_fc[b, 0:70] = inputs @ W_fc^T + b_fc, via WMMA f32 16x16x4.
// One wave per (16-row batch tile, 16-col out tile). K loop over C in steps 4.
//   A-frag (16x4): lane m = lane&15 supplies K = kq, kq+1 (kq = (lane>>4)*2)
//   B-frag (4x16): lane n = lane&15 supplies same K pair (B[k][n]=W_fc[n][k])
//   D (16x16 f32, 8 VGPRs): elem v -> row v + 8*(lane>=16), col lane&15
// Bounds on the 70->80 pad are handled branch-free (clamped row + 0/1 scale)
// so the inner loop is straight-line: loads pipeline ahead of the WMMAs and
// EXEC stays all-1s throughout (WMMA requirement).
// ---------------------------------------------------------------------------
__global__ __launch_bounds__(32)
void fc_wmma_kernel(const float* __restrict__ inp,
                    const float* __restrict__ Wfc,
                    const float* __restrict__ bfc,
                    float* __restrict__ out_fc) {
    const int lane  = threadIdx.x;
    const int m     = lane & 15;
    const int kq    = (lane >> 4) << 1;     // 0 or 2
    const int btile = blockIdx.x << 4;      // 0..511 in steps of 16
    const int otile = blockIdx.y << 4;      // 0,16,32,48,64

    const int   ocol   = otile + m;         // output column this lane feeds (B)
    const bool  ovld   = (ocol < (W_ + 6));
    const float bscale = ovld ? 1.f : 0.f;  // branch-free zeroing of pad cols
    const float* arow  = inp + (size_t)(btile + m) * C_;
    const float* brow  = Wfc + (size_t)(ovld ? ocol : 0) * C_;  // clamped, safe

    v8f acc = {};
#pragma unroll 8
    for (int c0 = 0; c0 < C_; c0 += 4) {
        v2f a = { arow[c0 + kq], arow[c0 + kq + 1] };
        v2f b = { brow[c0 + kq] * bscale, brow[c0 + kq + 1] * bscale };
        acc = __builtin_amdgcn_wmma_f32_16x16x4_f32(
                  false, a, false, b, (short)0, acc, false, false);
    }

    const int ncol = otile + (lane & 15);
    const int radd = (lane >> 4) << 3;      // +8 rows for upper half-wave
    const float bias = (ncol < (W_ + 6)) ? bfc[ncol] : 0.f;
#pragma unroll
    for (int v = 0; v < 8; ++v) {
        const int row = btile + v + radd;
        const float val = (ncol < (W_ + 6)) ? (acc[v] + bias) : 0.f;
        out_fc[(size_t)row * OPAD + ncol] = val;
    }
}

// ---------------------------------------------------------------------------
// Kernel 2: per-batch head parameters from FC output.
// params[b*8 + {0:beta,1:g,2:s0,3:s1,4:s2,5:gamma,6:knorm}]
// ---------------------------------------------------------------------------
__global__ __launch_bounds__(256)
void head_params_kernel(const float* __restrict__ out_fc,
                        float* __restrict__ params) {
    const int b = blockIdx.x * blockDim.x + threadIdx.x;
    if (b >= B_) return;
    const float* row = out_fc + (size_t)b * OPAD;

    const float beta  = softplus_f(row[W_ + 0]);
    const float g     = 1.f / (1.f + expf(-row[W_ + 1]));
    const float s0i = row[W_ + 2], s1i = row[W_ + 3], s2i = row[W_ + 4];
    const float mx = fmaxf(s0i, fmaxf(s1i, s2i));
    const float e0 = expf(s0i - mx), e1 = expf(s1i - mx), e2 = expf(s2i - mx);
    const float es = e0 + e1 + e2;
    const float gamma = 1.f + softplus_f(row[W_ + 5]);

    float sq = 0.f;
#pragma unroll 8
    for (int w = 0; w < W_; ++w) { const float kv = row[w]; sq += kv * kv; }

    float* p = params + (size_t)b * 8;
    p[0] = beta; p[1] = g; p[2] = e0 / es; p[3] = e1 / es; p[4] = e2 / es;
    p[5] = gamma; p[6] = sqrtf(sq); p[7] = 0.f;
}

// ---------------------------------------------------------------------------
// Kernel 3 (streaming pass 1 over M, 512 MiB): fused dot + row-norm + sim.
// abuf[b,n] = beta * dot / (||M[b,n]|| * ||k|| + EPS)   (abuf = output w slot)
// ---------------------------------------------------------------------------
__global__ __launch_bounds__(256)
void similarity_kernel(const float* __restrict__ M,
                       const float* __restrict__ out_fc,
                       const float* __restrict__ params,
                       float* __restrict__ abuf) {
    __shared__ float ks[W_];
    __shared__ float sbeta, sknorm;
    const int b = blockIdx.y;
    const int n = blockIdx.x * blockDim.x + threadIdx.x;

    if (threadIdx.x < W_) ks[threadIdx.x] = out_fc[(size_t)b * OPAD + threadIdx.x];
    if (threadIdx.x == 0) { sbeta = params[b * 8 + 0]; sknorm = params[b * 8 + 6]; }
    __syncthreads();

    const float4* mp = (const float4*)(M + ((size_t)b * N_ + n) * W_);
    __builtin_prefetch(mp + 8, 0, 0);   // global_prefetch_b8: 2nd half of row

    float dot = 0.f, sq = 0.f;
#pragma unroll
    for (int q = 0; q < 16; ++q) {
        const float4 mv = mp[q];
        dot += mv.x * ks[4 * q + 0] + mv.y * ks[4 * q + 1]
             + mv.z * ks[4 * q + 2] + mv.w * ks[4 * q + 3];
        sq  += mv.x * mv.x + mv.y * mv.y + mv.z * mv.z + mv.w * mv.w;
    }
    const float sim = dot / (sqrtf(sq) * sknorm + EPS_);
    abuf[(size_t)b * N_ + n] = sbeta * sim;
}

// ---------------------------------------------------------------------------
// Kernel 4: per-batch addressing — softmax over N, interpolate with w_pre,
// circular shift, sharpen (pow gamma), renormalize. In-place on wbuf row.
// One block per b; 4096-float row staged in LDS (17 KB / 320 KB WGP pool).
// ---------------------------------------------------------------------------
__global__ __launch_bounds__(256)
void address_kernel(const float* __restrict__ w_pre,
                    const float* __restrict__ params,
                    float* __restrict__ wbuf) {
    __shared__ float sh[N_];
    __shared__ float red[256];
    const int b = blockIdx.x, t = threadIdx.x;
    const float* pp = params + (size_t)b * 8;
    const float g = pp[1], s0 = pp[2], s1 = pp[3], s2 = pp[4], gamma = pp[5];
    float* wrow = wbuf + (size_t)b * N_;
    const float* prerow = w_pre + (size_t)b * N_;

    // 1) row max
    float mx = -3.0e38f;
    for (int n = t; n < N_; n += 256) { const float a = wrow[n]; sh[n] = a; mx = fmaxf(mx, a); }
    red[t] = mx; __syncthreads();
    for (int s = 128; s > 0; s >>= 1) { if (t < s) red[t] = fmaxf(red[t], red[t + s]); __syncthreads(); }
    mx = red[0]; __syncthreads();

    // 2) exp + sum
    float sum = 0.f;
    for (int n = t; n < N_; n += 256) { const float e = expf(sh[n] - mx); sh[n] = e; sum += e; }
    red[t] = sum; __syncthreads();
    for (int s = 128; s > 0; s >>= 1) { if (t < s) red[t] += red[t + s]; __syncthreads(); }
    sum = red[0]; __syncthreads();
    const float inv = 1.f / sum;

    // 3) w_g = g*w_c + (1-g)*w_pre  (each thread owns its own stripes of sh)
    for (int n = t; n < N_; n += 256) {
        sh[n] = g * (sh[n] * inv) + (1.f - g) * prerow[n];
    }
    __syncthreads();

    // 4) circular shift blend + sharpen
    float wp[16];
    float spow = 0.f;
#pragma unroll
    for (int i = 0; i < 16; ++i) {
        const int n = t + i * 256;
        const float wt = s0 * sh[(n + 1) & (N_ - 1)] + s1 * sh[n]
                       + s2 * sh[(n - 1) & (N_ - 1)];
        const float v = powf(wt, gamma);
        wp[i] = v; spow += v;
    }
    red[t] = spow; __syncthreads();
    for (int s = 128; s > 0; s >>= 1) { if (t < s) red[t] += red[t + s]; __syncthreads(); }
    const float invp = 1.f / (red[0] + EPS_);
#pragma unroll
    for (int i = 0; i < 16; ++i) wrow[t + i * 256] = wp[i] * invp;
}

// ---------------------------------------------------------------------------
// Kernel 5 (streaming pass 2 over M, 512 MiB): r[b,:] = sum_n w[b,n]*M[b,n,:].
// Lanes map to contiguous W dim -> every wave load is a contiguous 128 B line.
// ---------------------------------------------------------------------------
__global__ __launch_bounds__(256)
void readout_kernel(const float* __restrict__ M,
                    const float* __restrict__ wbuf,
                    float* __restrict__ rout) {
    __shared__ float wsh[N_];
    __shared__ float red[256];
    const int b = blockIdx.x, t = threadIdx.x;
    for (int n = t; n < N_; n += 256) wsh[n] = wbuf[(size_t)b * N_ + n];
    __syncthreads();

    const int wc  = t & (W_ - 1);
    const int grp = t >> 6;                 // 0..3 partial sums over n
    const float* mb = M + (size_t)b * N_ * W_;
    float acc = 0.f;
#pragma unroll 4
    for (int n = grp; n < N_; n += 4) {
        acc += wsh[n] * mb[(size_t)n * W_ + wc];
    }
    red[t] = acc; __syncthreads();
    if (t < W_) {
        rout[(size_t)b * W_ + t] = red[t] + red[t + 64] + red[t + 128] + red[t + 192];
    }
}

// ---------------------------------------------------------------------------
extern "C" void kernel_launch(void* const* d_in, const int* in_sizes, int n_in,
                              void* d_out, int out_size, void* d_ws, size_t ws_size,
                              hipStream_t stream) {
    const float* inputs = (const float*)d_in[0];   // (B, C)
    const float* w_pre  = (const float*)d_in[1];   // (B, N)
    const float* memM   = (const float*)d_in[2];   // (B, N, W)
    const float* Wfc    = (const float*)d_in[3];   // (W+6, C)
    const float* bfc    = (const float*)d_in[4];   // (W+6,)

    float* out  = (float*)d_out;
    float* rout = out;                      // r: B*W floats
    float* wbuf = out + (size_t)B_ * W_;    // w: B*N floats (also used as scratch)

    float* out_fc = (float*)d_ws;                       // B*OPAD floats
    float* params = out_fc + (size_t)B_ * OPAD;         // B*8 floats
    // total ws need: (512*80 + 512*8)*4 = 176 KB

    fc_wmma_kernel<<<dim3(B_ / 16, OPAD / 16), 32, 0, stream>>>(inputs, Wfc, bfc, out_fc);
    head_params_kernel<<<B_ / 256, 256, 0, stream>>>(out_fc, params);
    similarity_kernel<<<dim3(N_ / 256, B_), 256, 0, stream>>>(memM, out_fc, params, wbuf);
    address_kernel<<<B_, 256, 0, stream>>>(w_pre, params, wbuf);
    readout_kernel<<<B_, 256, 0, stream>>>(memM, wbuf, rout);
}